// FTTransformerPNAParallelLayer_83588653515398
// MI455X (gfx1250) — compile-verified
//
#include <hip/hip_runtime.h>
#include <math.h>

typedef __attribute__((ext_vector_type(2))) float v2f;
typedef __attribute__((ext_vector_type(8))) float v8f;

__device__ __forceinline__ v8f wmma4(v2f a, v2f b, v8f c) {
    // V_WMMA_F32_16X16X4_F32: fp32 matrix path (D = A(16x4) * B(4x16) + C)
    return __builtin_amdgcn_wmma_f32_16x16x4_f32(false, a, false, b, (short)0, c, false, false);
}

__device__ __forceinline__ unsigned fkey(float x) {
    unsigned b = __float_as_uint(x);
    return (b & 0x80000000u) ? ~b : (b | 0x80000000u);
}
__device__ __forceinline__ float fdec(unsigned k) {
    unsigned b = (k & 0x80000000u) ? (k ^ 0x80000000u) : ~k;
    return __uint_as_float(b);
}

// ---------------------------------------------------------------------------
// Generic GEMM: D[M,N] = A[M,K(lda)] @ W[N,K(ldb)]^T (+bias) (+res) (+relu)
// Block = 256 threads = 8 waves (4 M-tiles x 2 N-tiles), wave tile = 16x32.
// Requires: M % 16 == 0, N % 64 == 0, K % 4 == 0.
// ---------------------------------------------------------------------------
template <bool RELU>
__global__ __launch_bounds__(256) void gemm_abT(
    const float* __restrict__ A, const float* __restrict__ W,
    const float* __restrict__ bias, const float* __restrict__ res,
    float* __restrict__ D, int M, int N, int K, int lda, int ldb, int ldd)
{
    const int wave = threadIdx.x >> 5, lane = threadIdx.x & 31;
    const int wm = wave >> 1, wn = wave & 1;
    const int m0 = blockIdx.y * 64 + wm * 16;
    const int n0 = blockIdx.x * 64 + wn * 32;
    if (m0 >= M) return;                      // uniform per wave -> EXEC stays all-1 for WMMA
    const int lh = lane >> 4, l16 = lane & 15;

    const float* ap  = A + (size_t)(m0 + l16) * lda + 2 * lh;
    const float* bp0 = W + (size_t)(n0 + l16) * ldb + 2 * lh;
    const float* bp1 = bp0 + (size_t)16 * ldb;

    v8f acc0 = {}, acc1 = {};
    for (int k = 0; k < K; k += 4) {
        v2f a  = *(const v2f*)(ap + k);
        v2f b0 = *(const v2f*)(bp0 + k);
        v2f b1 = *(const v2f*)(bp1 + k);
        acc0 = wmma4(a, b0, acc0);
        acc1 = wmma4(a, b1, acc1);
    }
    const int c0 = n0 + l16, c1 = c0 + 16;
    const float bi0 = bias ? bias[c0] : 0.f;
    const float bi1 = bias ? bias[c1] : 0.f;
    for (int r = 0; r < 8; ++r) {
        int row = m0 + 8 * lh + r;
        float v0 = acc0[r] + bi0;
        float v1 = acc1[r] + bi1;
        if (res) {
            v0 += res[(size_t)row * ldd + c0];
            v1 += res[(size_t)row * ldd + c1];
        }
        if (RELU) { v0 = fmaxf(v0, 0.f); v1 = fmaxf(v1, 0.f); }
        D[(size_t)row * ldd + c0] = v0;
        D[(size_t)row * ldd + c1] = v1;
    }
}

// ---------------------------------------------------------------------------
// Multi-head attention for one (b,h): scores=q@k^T/sqrt(dh), softmax, o=attn@v
// B=1024,S=64,C=256,H=8,dh=32. One block (8 waves) per (b,h).
// ---------------------------------------------------------------------------
__global__ __launch_bounds__(256) void attn_kernel(
    const float* __restrict__ qkv, float* __restrict__ out /* [B*S, 256] */)
{
    __shared__ float sm[64 * 66];
    const int bh = blockIdx.x;
    const int b = bh >> 3, h = bh & 7;
    const int wave = threadIdx.x >> 5, lane = threadIdx.x & 31;
    const int lh = lane >> 4, l16 = lane & 15;
    const float* base = qkv + (size_t)b * 64 * 768;
    const float* qb = base + h * 32;
    const float* kb = base + 256 + h * 32;
    const float* vb = base + 512 + h * 32;

    // scores (64x64, K=32): waves 4(M)x2(N), wave tile 16x32
    {
        const int wm = wave >> 1, wn = wave & 1;
        const int m0 = wm * 16, n0 = wn * 32;
        const float* ap  = qb + (size_t)(m0 + l16) * 768 + 2 * lh;
        const float* bp0 = kb + (size_t)(n0 + l16) * 768 + 2 * lh;
        const float* bp1 = bp0 + (size_t)16 * 768;
        v8f acc0 = {}, acc1 = {};
        for (int k = 0; k < 32; k += 4) {
            v2f a  = *(const v2f*)(ap + k);
            v2f b0 = *(const v2f*)(bp0 + k);
            v2f b1 = *(const v2f*)(bp1 + k);
            acc0 = wmma4(a, b0, acc0);
            acc1 = wmma4(a, b1, acc1);
        }
        const float scale = 0.17677669529663687f;  // 1/sqrt(32)
        for (int r = 0; r < 8; ++r) {
            int row = m0 + 8 * lh + r;
            sm[row * 66 + n0 + l16]      = acc0[r] * scale;
            sm[row * 66 + n0 + 16 + l16] = acc1[r] * scale;
        }
    }
    __syncthreads();
    if (threadIdx.x < 64) {                 // row softmax
        float* rp = &sm[threadIdx.x * 66];
        float mx = rp[0];
        for (int c = 1; c < 64; ++c) mx = fmaxf(mx, rp[c]);
        float s = 0.f;
        for (int c = 0; c < 64; ++c) { float e = __expf(rp[c] - mx); rp[c] = e; s += e; }
        float inv = 1.f / s;
        for (int c = 0; c < 64; ++c) rp[c] *= inv;
    }
    __syncthreads();
    // o = attn @ v (64x32, K=64): 8 waves, one 16x16 tile each
    {
        const int wm = wave >> 1, wn = wave & 1;
        const int m0 = wm * 16, n0 = wn * 16;
        v8f acc = {};
        for (int k = 0; k < 64; k += 4) {
            v2f a = *(const v2f*)(&sm[(m0 + l16) * 66 + k + 2 * lh]);
            int kk = k + 2 * lh;
            v2f bb;
            bb.x = vb[(size_t)(kk)     * 768 + n0 + l16];   // B[k][n] = v[k][n]
            bb.y = vb[(size_t)(kk + 1) * 768 + n0 + l16];
            acc = wmma4(a, bb, acc);
        }
        for (int r = 0; r < 8; ++r) {
            int row = b * 64 + m0 + 8 * lh + r;
            out[(size_t)row * 256 + h * 32 + n0 + l16] = acc[r];
        }
    }
}

// LayerNorm over C=256 per row; one 256-thread block per row.
__global__ __launch_bounds__(256) void ln_kernel(
    const float* __restrict__ in, const float* __restrict__ g,
    const float* __restrict__ bta, float* __restrict__ out)
{
    __shared__ float red[256];
    const int row = blockIdx.x, t = threadIdx.x;
    float x = in[(size_t)row * 256 + t];
    red[t] = x; __syncthreads();
    for (int s = 128; s > 0; s >>= 1) { if (t < s) red[t] += red[t + s]; __syncthreads(); }
    float mu = red[0] * (1.f / 256.f); __syncthreads();
    float d = x - mu;
    red[t] = d * d; __syncthreads();
    for (int s = 128; s > 0; s >>= 1) { if (t < s) red[t] += red[t + s]; __syncthreads(); }
    float rstd = rsqrtf(red[0] * (1.f / 256.f) + 1e-5f);
    out[(size_t)row * 256 + t] = g[t] * d * rstd + bta[t];
}

// pre_nn GEMM with fused gather: A row e = [x[dst[e]] | x[src[e]] | ebuf[e]], K=384, N=128
__global__ __launch_bounds__(256) void gemm_pre(
    const float* __restrict__ xg, const float* __restrict__ ebuf,
    const int* __restrict__ src, const int* __restrict__ dst,
    const float* __restrict__ W, const float* __restrict__ bias,
    float* __restrict__ D, int E)
{
    const int wave = threadIdx.x >> 5, lane = threadIdx.x & 31;
    const int wm = wave >> 1, wn = wave & 1;
    const int m0 = blockIdx.y * 64 + wm * 16;
    const int n0 = blockIdx.x * 64 + wn * 32;
    if (m0 >= E) return;
    const int lh = lane >> 4, l16 = lane & 15;
    const int e = m0 + l16;
    const float* pa0 = xg + (size_t)dst[e] * 128 + 2 * lh;
    const float* pa1 = xg + (size_t)src[e] * 128 + 2 * lh;
    const float* pa2 = ebuf + (size_t)e * 128 + 2 * lh;
    const float* bp0 = W + (size_t)(n0 + l16) * 384 + 2 * lh;
    const float* bp1 = bp0 + (size_t)16 * 384;
    v8f acc0 = {}, acc1 = {};
    const float* segs[3] = { pa0, pa1, pa2 };
    for (int seg = 0; seg < 3; ++seg) {
        const float* ap = segs[seg];
        const float* wp0 = bp0 + seg * 128;
        const float* wp1 = bp1 + seg * 128;
        for (int k = 0; k < 128; k += 4) {
            v2f a  = *(const v2f*)(ap + k);
            v2f b0 = *(const v2f*)(wp0 + k);
            v2f b1 = *(const v2f*)(wp1 + k);
            acc0 = wmma4(a, b0, acc0);
            acc1 = wmma4(a, b1, acc1);
        }
    }
    const int c0 = n0 + l16, c1 = c0 + 16;
    const float bi0 = bias[c0], bi1 = bias[c1];
    for (int r = 0; r < 8; ++r) {
        int row = m0 + 8 * lh + r;
        D[(size_t)row * 128 + c0] = acc0[r] + bi0;
        D[(size_t)row * 128 + c1] = acc1[r] + bi1;
    }
}

// Init stats buffer [N,512]: cols 0-127 sum(0), 128-255 maxkey(0), 256-383 minkey(~0), 384-511 sum2(0)
__global__ void gnn_init(unsigned* __restrict__ S4u, float* __restrict__ cnt,
                         float* __restrict__ bns, float* __restrict__ bns2, int N)
{
    size_t idx = (size_t)blockIdx.x * blockDim.x + threadIdx.x;
    size_t tot = (size_t)N * 512;
    if (idx < tot) {
        int f = (int)(idx & 511);
        S4u[idx] = (f >= 256 && f < 384) ? 0xFFFFFFFFu : 0u;
    }
    if (idx < (size_t)N) cnt[idx] = 0.f;
    if (idx < 128) { bns[idx] = 0.f; bns2[idx] = 0.f; }
}

__global__ void agg_kernel(const float* __restrict__ h, const int* __restrict__ dst,
                           float* __restrict__ S4, unsigned* __restrict__ S4u,
                           float* __restrict__ cnt, int E)
{
    size_t idx = (size_t)blockIdx.x * blockDim.x + threadIdx.x;
    if (idx >= (size_t)E * 128) return;
    int e = (int)(idx >> 7), f = (int)(idx & 127);
    float v = h[idx];
    size_t base = (size_t)dst[e] * 512 + f;
    atomicAdd(&S4[base], v);
    atomicAdd(&S4[base + 384], v * v);
    unsigned k = fkey(v);
    atomicMax(&S4u[base + 128], k);
    atomicMin(&S4u[base + 256], k);
    if (f == 0) atomicAdd(&cnt[dst[e]], 1.f);
}

__global__ void combine_kernel(float* __restrict__ S4, unsigned* __restrict__ S4u,
                               const float* __restrict__ cnt, int N)
{
    size_t idx = (size_t)blockIdx.x * blockDim.x + threadIdx.x;
    if (idx >= (size_t)N * 128) return;
    int n = (int)(idx >> 7), f = (int)(idx & 127);
    float c = cnt[n];
    float denom = fmaxf(c, 1.f);
    size_t base = (size_t)n * 512 + f;
    float mean = S4[base] / denom;
    float m2   = S4[base + 384] / denom;
    float stdv = sqrtf(fmaxf(m2 - mean * mean, 0.f) + 1e-5f);
    float mx = 0.f, mn = 0.f;
    if (c > 0.f) { mx = fdec(S4u[base + 128]); mn = fdec(S4u[base + 256]); }
    S4[base]       = mean;
    S4[base + 128] = mx;
    S4[base + 256] = mn;
    S4[base + 384] = stdv;
}

// post_nn folded: out = x@Wx^T + agg@Wa^T + amp*(agg@Wb^T) + att*(agg@Wc^T) + b
// W = post_w [128, 1664]; Wx cols 0:128, Wa 128:640, Wb 640:1152, Wc 1152:1664
__global__ __launch_bounds__(256) void gemm_post(
    const float* __restrict__ xg, const float* __restrict__ S4,
    const float* __restrict__ cnt, const float* __restrict__ W,
    const float* __restrict__ bias, float* __restrict__ D, int N)
{
    const int wave = threadIdx.x >> 5, lane = threadIdx.x & 31;
    const int wm = wave >> 1, wn = wave & 1;
    const int m0 = blockIdx.y * 64 + wm * 16;
    const int n0 = blockIdx.x * 64 + wn * 32;
    if (m0 >= N) return;
    const int lh = lane >> 4, l16 = lane & 15;
    const float* axp = xg + (size_t)(m0 + l16) * 128 + 2 * lh;
    const float* asp = S4 + (size_t)(m0 + l16) * 512 + 2 * lh;
    const float* wb0 = W + (size_t)(n0 + l16) * 1664 + 2 * lh;
    const float* wb1 = wb0 + (size_t)16 * 1664;
    v8f ax0 = {}, ax1 = {}, aa0 = {}, aa1 = {}, ab0 = {}, ab1 = {}, ac0 = {}, ac1 = {};
    for (int k = 0; k < 128; k += 4) {
        v2f a  = *(const v2f*)(axp + k);
        v2f b0 = *(const v2f*)(wb0 + k);
        v2f b1 = *(const v2f*)(wb1 + k);
        ax0 = wmma4(a, b0, ax0); ax1 = wmma4(a, b1, ax1);
    }
    for (int k = 0; k < 512; k += 4) {
        v2f a = *(const v2f*)(asp + k);
        v2f ba0 = *(const v2f*)(wb0 + 128 + k);
        v2f ba1 = *(const v2f*)(wb1 + 128 + k);
        aa0 = wmma4(a, ba0, aa0); aa1 = wmma4(a, ba1, aa1);
        v2f bb0 = *(const v2f*)(wb0 + 640 + k);
        v2f bb1 = *(const v2f*)(wb1 + 640 + k);
        ab0 = wmma4(a, bb0, ab0); ab1 = wmma4(a, bb1, ab1);
        v2f bc0 = *(const v2f*)(wb0 + 1152 + k);
        v2f bc1 = *(const v2f*)(wb1 + 1152 + k);
        ac0 = wmma4(a, bc0, ac0); ac1 = wmma4(a, bc1, ac1);
    }
    const int c0 = n0 + l16, c1 = c0 + 16;
    const float DELTA = 1.2059890135352093f;
    const float bi0 = bias[c0], bi1 = bias[c1];
    for (int r = 0; r < 8; ++r) {
        int row = m0 + 8 * lh + r;
        float deg = fmaxf(cnt[row], 1.f);
        float lg = logf(deg + 1.f);
        float amp = lg / DELTA, att = DELTA / lg;
        D[(size_t)row * 128 + c0] = ax0[r] + aa0[r] + amp * ab0[r] + att * ac0[r] + bi0;
        D[(size_t)row * 128 + c1] = ax1[r] + aa1[r] + amp * ab1[r] + att * ac1[r] + bi1;
    }
}

__global__ __launch_bounds__(128) void bn_stats(
    const float* __restrict__ X, float* __restrict__ bns, float* __restrict__ bns2, int N)
{
    int f = threadIdx.x;
    float s = 0.f, s2 = 0.f;
    for (int r = blockIdx.x; r < N; r += gridDim.x) {
        float v = X[(size_t)r * 128 + f];
        s += v; s2 += v * v;
    }
    atomicAdd(&bns[f], s);
    atomicAdd(&bns2[f], s2);
}

__global__ void gnn_out(const float* __restrict__ X, const float* __restrict__ xg,
                        const float* __restrict__ bns, const float* __restrict__ bns2,
                        const float* __restrict__ g, const float* __restrict__ b,
                        float* __restrict__ out, int N)
{
    size_t idx = (size_t)blockIdx.x * blockDim.x + threadIdx.x;
    if (idx >= (size_t)N * 128) return;
    int f = (int)(idx & 127);
    float invN = 1.f / (float)N;
    float mu = bns[f] * invN;
    float var = fmaxf(bns2[f] * invN - mu * mu, 0.f);
    float y = g[f] * (X[idx] - mu) * rsqrtf(var + 1e-5f) + b[f];
    out[idx] = 0.5f * (xg[idx] + fmaxf(y, 0.f));
}

__global__ void copy_kernel(const float* __restrict__ s, float* __restrict__ d, size_t n)
{
    size_t i = (size_t)blockIdx.x * blockDim.x + threadIdx.x;
    if (i < n) d[i] = s[i];
}

// ---------------------------------------------------------------------------
extern "C" void kernel_launch(void* const* d_in, const int* in_sizes, int n_in,
                              void* d_out, int out_size, void* d_ws, size_t ws_size,
                              hipStream_t stream)
{
    const float* x_tab      = (const float*)d_in[0];
    const float* x_gnn      = (const float*)d_in[1];
    const int*   eidx       = (const int*)  d_in[2];
    const float* eattr      = (const float*)d_in[3];
    const float* in_proj_w  = (const float*)d_in[4];
    const float* in_proj_b  = (const float*)d_in[5];
    const float* out_proj_w = (const float*)d_in[6];
    const float* out_proj_b = (const float*)d_in[7];
    const float* lin1_w     = (const float*)d_in[8];
    const float* lin1_b     = (const float*)d_in[9];
    const float* lin2_w     = (const float*)d_in[10];
    const float* lin2_b     = (const float*)d_in[11];
    const float* ln1_g      = (const float*)d_in[12];
    const float* ln1_b      = (const float*)d_in[13];
    const float* ln2_g      = (const float*)d_in[14];
    const float* ln2_b      = (const float*)d_in[15];
    const float* tabn_g     = (const float*)d_in[16];
    const float* tabn_b     = (const float*)d_in[17];
    const float* enc_w      = (const float*)d_in[18];
    const float* enc_b      = (const float*)d_in[19];
    const float* pre_w      = (const float*)d_in[20];
    const float* pre_b      = (const float*)d_in[21];
    const float* post_w     = (const float*)d_in[22];
    const float* post_b     = (const float*)d_in[23];
    const float* linp_w     = (const float*)d_in[24];
    const float* linp_b     = (const float*)d_in[25];
    const float* bn_g       = (const float*)d_in[26];
    const float* bn_b       = (const float*)d_in[27];
    (void)in_sizes; (void)n_in; (void)out_size; (void)ws_size;

    constexpr int Bt = 1024, C = 256, H = 8, FFd = 1024;
    constexpr int Nn = 100000, Ee = 400000, F = 128;
    constexpr int Mt = Bt * 64;   // 65536 rows

    float* ws  = (float*)d_ws;
    float* R0  = ws;                 // 67,108,864 floats (ffn / ebuf / S4)
    float* R1  = R0 + 67108864;      // 51,200,000 floats (qkv / h1 / hbuf)
    float* R2  = R1 + 51200000;      // 16,777,216 floats (attn_o / s2 / pna1)
    float* R3  = R2 + 16777216;      // 16,777,216 floats (s1 / h2 / pna2)
    float* cnt = R3 + 16777216;      // 100,000
    float* bns = cnt + 100000;       // 128
    float* bns2 = bns + 128;         // 128

    const int* src = eidx;           // edge_index[0] = j (source)
    const int* dst = eidx + Ee;      // edge_index[1] = i (aggregate at dst)

    float* d_tab = (float*)d_out;                    // Mt*C
    float* d_gnn = d_tab + (size_t)Mt * C;           // Nn*F
    float* d_ea  = d_gnn + (size_t)Nn * F;           // Ee*F

    dim3 blk(256);

    // ---------------- tab branch ----------------
    float* qkv = R1;
    gemm_abT<false><<<dim3(768 / 64, Mt / 64), blk, 0, stream>>>(
        x_tab, in_proj_w, in_proj_b, nullptr, qkv, Mt, 3 * C, C, C, C, 3 * C);
    float* attn_o = R2;
    attn_kernel<<<dim3(Bt * H), blk, 0, stream>>>(qkv, attn_o);
    float* s1 = R3;
    gemm_abT<false><<<dim3(C / 64, Mt / 64), blk, 0, stream>>>(
        attn_o, out_proj_w, out_proj_b, x_tab, s1, Mt, C, C, C, C, C);
    float* h1 = R1;
    ln_kernel<<<dim3(Mt), blk, 0, stream>>>(s1, ln1_g, ln1_b, h1);
    float* ffn = R0;
    gemm_abT<true><<<dim3(FFd / 64, Mt / 64), blk, 0, stream>>>(
        h1, lin1_w, lin1_b, nullptr, ffn, Mt, FFd, C, C, C, FFd);
    float* s2 = R2;
    gemm_abT<false><<<dim3(C / 64, Mt / 64), blk, 0, stream>>>(
        ffn, lin2_w, lin2_b, h1, s2, Mt, C, FFd, FFd, FFd, C);
    float* h2 = R3;
    ln_kernel<<<dim3(Mt), blk, 0, stream>>>(s2, ln2_g, ln2_b, h2);
    ln_kernel<<<dim3(Mt), blk, 0, stream>>>(h2, tabn_g, tabn_b, d_tab);

    // ---------------- gnn branch ----------------
    float* ebuf = R0;
    gemm_abT<false><<<dim3(F / 64, Ee / 64), blk, 0, stream>>>(
        eattr, enc_w, enc_b, nullptr, ebuf, Ee, F, F, F, F, F);
    float* hbuf = R1;
    gemm_pre<<<dim3(F / 64, Ee / 64), blk, 0, stream>>>(
        x_gnn, ebuf, src, dst, pre_w, pre_b, hbuf, Ee);
    float* S4 = R0;                    // ebuf dead after gemm_pre
    unsigned* S4u = (unsigned*)S4;
    {
        unsigned nb = (unsigned)(((size_t)Nn * 512 + 255) / 256);
        gnn_init<<<dim3(nb), blk, 0, stream>>>(S4u, cnt, bns, bns2, Nn);
    }
    agg_kernel<<<dim3((unsigned)(((size_t)Ee * F + 255) / 256)), blk, 0, stream>>>(
        hbuf, dst, S4, S4u, cnt, Ee);
    combine_kernel<<<dim3((unsigned)(((size_t)Nn * F + 255) / 256)), blk, 0, stream>>>(
        S4, S4u, cnt, Nn);
    float* pna1 = R2;
    gemm_post<<<dim3(F / 64, (Nn + 63) / 64), blk, 0, stream>>>(
        x_gnn, S4, cnt, post_w, post_b, pna1, Nn);
    float* pna2 = R3;
    gemm_abT<false><<<dim3(F / 64, (Nn + 63) / 64), blk, 0, stream>>>(
        pna1, linp_w, linp_b, nullptr, pna2, Nn, F, F, F, F, F);
    bn_stats<<<dim3(512), dim3(128), 0, stream>>>(pna2, bns, bns2, Nn);
    gnn_out<<<dim3((unsigned)(((size_t)Nn * F + 255) / 256)), blk, 0, stream>>>(
        pna2, x_gnn, bns, bns2, bn_g, bn_b, d_gnn, Nn);
    copy_kernel<<<dim3((unsigned)(((size_t)Ee * F + 255) / 256)), blk, 0, stream>>>(
        eattr, d_ea, (size_t)Ee * F);
}